// BackProjectionLinear_23983097381102
// MI455X (gfx1250) — compile-verified
//
#include <hip/hip_runtime.h>
#include <hip/hip_bf16.h>
#include <math.h>

// CDNA5 / gfx1250: wave32, WMMA f32 16x16x4 for the detector reduction.
typedef __attribute__((ext_vector_type(2))) float v2f;
typedef __attribute__((ext_vector_type(8))) float v8f;

#define NB   4
#define NDET 128
#define NT   2048
#define NPIX (256 * 256)
#define WAVES_PER_BLOCK 8

__global__ __launch_bounds__(256)
void bp_wmma_f32_kernel(const float* __restrict__ sino,   // [4][128][2048]
                        const float* __restrict__ lut,    // [256][256][128][2] = (k_floor, alpha)
                        float* __restrict__ out)          // [4][1][256][256]
{
    // Hann apodization window, computed once per block (cosf stays OUT of the hot loop).
    __shared__ float sApod[NDET];
    if (threadIdx.x < NDET) {
        const float w = 6.28318530717958647692f / 127.0f;  // 2*pi / (N_DET-1)
        sApod[threadIdx.x] = 0.5f - 0.5f * cosf((float)threadIdx.x * w);
    }
    __syncthreads();

    const int lane = threadIdx.x & 31;
    const int wave = threadIdx.x >> 5;
    const int tile = blockIdx.x * WAVES_PER_BLOCK + wave;  // one wave == 16 pixels
    const int p0   = tile << 4;
    const int halfSel = (lane < 16) ? 0 : 2;               // A-matrix K split (ISA 16x4 f32 layout)
    const int prow = p0 + (lane & 15);                     // this lane's pixel row (M index)

    // Accumulators: D/C matrix, 8 VGPRs each, one per batch.
    v8f acc[NB] = { v8f{0}, v8f{0}, v8f{0}, v8f{0} };

    for (int d = 0; d < NDET; d += 4) {
        const int det0 = d + halfSel;

        // LUT is contiguous in det per pixel: aligned float4 = (k0,a0,k1,a1) for dets det0,det0+1.
        const float4 L = *(const float4*)(lut + ((size_t)prow * NDET + det0) * 2);

        // B matrix (4x16): apod[d+k] broadcast across all 16 columns.
        // lanes 0-15 hold K-rows 0,1 ; lanes 16-31 hold K-rows 2,3.
        // det0 is even -> 8-byte aligned ds_load_b64.
        const float2 ap = *(const float2*)(&sApod[det0]);
        v2f bb; bb.x = ap.x; bb.y = ap.y;

        const int kf0 = (int)L.x;
        const int kf1 = (int)L.z;
        const bool valid0 = (kf0 >= 0) && (kf0 < NT - 1);
        const bool valid1 = (kf1 >= 0) && (kf1 < NT - 1);
        const int k00 = min(max(kf0, 0), NT - 2);
        const int k10 = min(max(kf1, 0), NT - 2);
        const int roff0 = det0 * NT + k00;          // offset within one batch's sino
        const int roff1 = (det0 + 1) * NT + k10;
        const float a0 = L.y, a1 = L.w;

#pragma unroll
        for (int b = 0; b < NB; ++b) {
            const float* s = sino + (size_t)b * (NDET * NT);
            // sino is 4 MiB -> L2-resident; gathers hit cache.
            const float s00 = s[roff0], s01 = s[roff0 + 1];
            const float s10 = s[roff1], s11 = s[roff1 + 1];
            float sk0 = (1.0f - a0) * s00 + a0 * s01;
            float sk1 = (1.0f - a1) * s10 + a1 * s11;
            sk0 = valid0 ? sk0 : 0.0f;
            sk1 = valid1 ? sk1 : 0.0f;

            v2f A; A.x = sk0; A.y = sk1;   // A 16x4: VGPR0=K even, VGPR1=K odd within half

            // D = A(16x4) * B(4x16) + C, full f32, RNE accumulate.
            acc[b] = __builtin_amdgcn_wmma_f32_16x16x4_f32(
                /*neg_a=*/false, A, /*neg_b=*/false, bb,
                /*c_mod=*/(short)0, acc[b], /*reuse_a=*/false, /*reuse_b=*/false);
        }
    }

    // norm = sum(apod) = 64 - 0.5*sum(cos(2*pi*d/127)) = 63.5 exactly (period-127 cosine sum).
    const float invNorm = 1.0f / 63.5f;

    // D layout: VGPR r, lane L -> row M = r + (L<16?0:8), col N = L%16. All columns identical.
    if (lane == 0 || lane == 16) {
        const int rbase = p0 + ((lane == 16) ? 8 : 0);
#pragma unroll
        for (int b = 0; b < NB; ++b) {
#pragma unroll
            for (int r = 0; r < 8; ++r) {
                out[(size_t)b * NPIX + rbase + r] = acc[b][r] * invNorm;
            }
        }
    }
}

extern "C" void kernel_launch(void* const* d_in, const int* in_sizes, int n_in,
                              void* d_out, int out_size, void* d_ws, size_t ws_size,
                              hipStream_t stream) {
    const float* sino = (const float*)d_in[0];  // (4,1,128,2048) f32
    const float* lut  = (const float*)d_in[1];  // (256,256,128,2) f32
    float* out = (float*)d_out;                 // (4,1,256,256) f32

    const int tiles  = NPIX / 16;                       // 4096 waves
    const int blocks = tiles / WAVES_PER_BLOCK;         // 512 blocks x 256 threads (8 waves)
    bp_wmma_f32_kernel<<<blocks, 256, 0, stream>>>(sino, lut, out);
}